// NetGrad_9517647528414
// MI455X (gfx1250) — compile-verified
//
#include <hip/hip_runtime.h>

// ---------------------------------------------------------------------------
// NetGrad Jacobian on MI455X (gfx1250, wave32, WMMA f32 16x16x4)
//
// grad = W5 D4 W4 D3 W3 D2 W2 D1 W1  (per sample), computed left-to-right:
//   forward:  g_i = elu'(pre_i), h_i = elu(pre_i)   (4 WMMA GEMMs, fused epi)
//   backward: U <- (U @ W) * g    with the 3 output rows stacked into M=24576
// Total ~14 GFLOP instead of the reference's ~206 GFLOP association.
//
// v2: k-pair-contiguous LDS layouts -> every WMMA operand is one aligned
//     ds_load_b64 (no v_mov gather); 2Mx4N wave tiling -> 6 ds loads per
//     8 WMMAs with B-fragment reuse across M tiles.
// ---------------------------------------------------------------------------

typedef __attribute__((ext_vector_type(2))) float v2f;
typedef __attribute__((ext_vector_type(4))) float v4f;
typedef __attribute__((ext_vector_type(8))) float v8f;

#define BM   64
#define KC   16
#define NSMP 8192          // batch size B
#define HDIM 256           // hidden width

// EPI: 0 = forward, store g and h
//      1 = forward, store g only
//      2 = backward, out = (A@W) * gscale   (in-place safe: BN == full N)
//      3 = final,    out[b*192 + r*64 + n] = (A@W)
template <int EPI, int BN>
__global__ __launch_bounds__(256) void gemm_wmma(
    const float* __restrict__ A, int lda,
    const float* __restrict__ W, int ldw,
    const float* __restrict__ bias,    // EPI 0/1
    const float* __restrict__ gscale,  // EPI 2 : [NSMP, N]
    float* __restrict__ outG,          // EPI 0/1
    float* __restrict__ outH,          // EPI 0
    float* __restrict__ outC,          // EPI 2/3
    int M, int N, int K)
{
    constexpr int NT  = BN / 16;          // N tiles per block
    constexpr int TPW = (4 * NT) / 8;     // C tiles per wave (8 or 2)
    constexpr int JM  = (TPW >= 4) ? 2 : 1;
    constexpr int JN  = TPW / JM;         // 4 or 2
    constexpr int MG  = 4 / JM;           // M wave-groups
    constexpr int F4  = BN / 64;          // float4 loads per thread for B tile
    constexpr int BSTR = 2 * BN + 32;     // pair-packed B row stride (banks)

    // A: row-major, stride KC+2 keeps b64 alignment + permutes banks.
    // B: element (k,n) at Bs[k>>1][2n + (k&1)]  -> frag = one b64 load.
    __shared__ __align__(16) float As[BM][KC + 2];
    __shared__ __align__(16) float Bs[KC / 2][BSTR];

    const int m0   = blockIdx.x * BM;
    const int n0   = blockIdx.y * BN;
    const int t    = threadIdx.x;
    const int wave = t >> 5;
    const int lane = t & 31;
    const int mgroup = wave % MG;
    const int ngroup = wave / MG;
    const int lrow   = lane & 15;
    const int khalf  = (lane >> 4) << 1;  // A/B frag K offset: 0 or 2
    const int chalf  = (lane >> 4) << 3;  // C frag M offset:   0 or 8

    v8f acc[JM][JN];
#pragma unroll
    for (int jm = 0; jm < JM; ++jm)
#pragma unroll
        for (int jn = 0; jn < JN; ++jn)
            acc[jm][jn] = (v8f){0.f,0.f,0.f,0.f,0.f,0.f,0.f,0.f};

    for (int kk = 0; kk < K; kk += KC) {
        // ---- stage A tile: BM x KC ----
        {
            const int r = t >> 2;            // 0..63
            const int c = (t & 3) << 2;      // 0,4,8,12
            v4f v = *(const v4f*)(A + (size_t)(m0 + r) * lda + kk + c);
            *(v2f*)&As[r][c + 0] = (v2f){v.x, v.y};
            *(v2f*)&As[r][c + 2] = (v2f){v.z, v.w};
            if (kk + KC < K)   // hint next A tile into cache (global_prefetch_b8)
                __builtin_prefetch(A + (size_t)(m0 + r) * lda + kk + KC + c, 0, 0);
        }
        // ---- stage B tile: KC x BN (pair-packed) ----
        if (EPI <= 1) {
            // forward: B[k][n] = W[(n0+n)*ldw + kk+k]  (W^T)
            const int n  = t % BN;
            const int cb = (t / BN) * (4 * F4);
#pragma unroll
            for (int e = 0; e < F4; ++e) {
                const int c = cb + e * 4;
                v4f v = *(const v4f*)(W + (size_t)(n0 + n) * ldw + kk + c);
                *(v2f*)&Bs[(c + 0) >> 1][2 * n] = (v2f){v.x, v.y};
                *(v2f*)&Bs[(c + 2) >> 1][2 * n] = (v2f){v.z, v.w};
            }
        } else {
            // backward: B[k][n] = W[(kk+k)*ldw + n0+n]
            const int k  = t >> 4;                   // 0..15
            const int cb = (t & 15) * (BN / 16);
            const int kp = k >> 1, ko = k & 1;
#pragma unroll
            for (int e = 0; e < F4; ++e) {
                const int c = cb + e * 4;
                v4f v = *(const v4f*)(W + (size_t)(kk + k) * ldw + n0 + c);
                Bs[kp][2 * (c + 0) + ko] = v.x;
                Bs[kp][2 * (c + 1) + ko] = v.y;
                Bs[kp][2 * (c + 2) + ko] = v.z;
                Bs[kp][2 * (c + 3) + ko] = v.w;
            }
        }
        __syncthreads();

        // ---- 4 WMMA k-steps per staged chunk ----
#pragma unroll
        for (int ks = 0; ks < KC; ks += 4) {
            const int kp = (ks + khalf) >> 1;
            v2f a[JM];
#pragma unroll
            for (int jm = 0; jm < JM; ++jm)
                a[jm] = *(const v2f*)&As[(mgroup * JM + jm) * 16 + lrow][ks + khalf];
#pragma unroll
            for (int jn = 0; jn < JN; ++jn) {
                const int nc = (ngroup * JN + jn) * 16 + lrow;
                const v2f b = *(const v2f*)&Bs[kp][2 * nc];
#pragma unroll
                for (int jm = 0; jm < JM; ++jm)
                    acc[jm][jn] = __builtin_amdgcn_wmma_f32_16x16x4_f32(
                        false, a[jm], false, b, (short)0, acc[jm][jn], false, false);
            }
        }
        __syncthreads();
    }

    // ---- epilogue ----
#pragma unroll
    for (int jm = 0; jm < JM; ++jm) {
#pragma unroll
        for (int jn = 0; jn < JN; ++jn) {
            const int ncol = n0 + (ngroup * JN + jn) * 16 + lrow;
#pragma unroll
            for (int v = 0; v < 8; ++v) {
                const int mrow = m0 + (mgroup * JM + jm) * 16 + chalf + v;
                float val = acc[jm][jn][v];
                if (EPI == 0 || EPI == 1) {
                    const float pre = val + bias[ncol];
                    const float e   = __expf(pre);
                    const bool  pos = pre > 0.0f;
                    outG[(size_t)mrow * N + ncol] = pos ? 1.0f : e;          // elu'
                    if (EPI == 0)
                        outH[(size_t)mrow * N + ncol] = pos ? pre : (e - 1.0f); // elu
                } else if (EPI == 2) {
                    const int b = mrow & (NSMP - 1);
                    outC[(size_t)mrow * N + ncol] =
                        val * gscale[(size_t)b * N + ncol];
                } else {  // EPI == 3: N == 64, scatter to [B, 3, 64]
                    const int b = mrow & (NSMP - 1);
                    const int r = mrow >> 13;
                    outC[(size_t)b * 192 + r * 64 + ncol] = val;
                }
            }
        }
    }
}

// U4[r*B + b][k] = W5[r][k] * g4[b][k]
__global__ __launch_bounds__(256) void init_u4(
    const float* __restrict__ W5, const float* __restrict__ g4,
    float* __restrict__ U)
{
    const int idx = blockIdx.x * 256 + threadIdx.x;  // over 3*B*256
    const int k = idx & (HDIM - 1);
    const int m = idx >> 8;
    const int b = m & (NSMP - 1);
    const int r = m >> 13;
    U[idx] = W5[r * HDIM + k] * g4[(size_t)b * HDIM + k];
}

extern "C" void kernel_launch(void* const* d_in, const int* in_sizes, int n_in,
                              void* d_out, int out_size, void* d_ws, size_t ws_size,
                              hipStream_t stream) {
    (void)in_sizes; (void)n_in; (void)out_size; (void)ws_size;
    const float* x  = (const float*)d_in[0];
    const float* W1 = (const float*)d_in[1];
    const float* b1 = (const float*)d_in[2];
    const float* W2 = (const float*)d_in[3];
    const float* b2 = (const float*)d_in[4];
    const float* W3 = (const float*)d_in[5];
    const float* b3 = (const float*)d_in[6];
    const float* W4 = (const float*)d_in[7];
    const float* b4 = (const float*)d_in[8];
    const float* W5 = (const float*)d_in[9];
    float* out = (float*)d_out;

    // workspace: g1..g4 [8192,256] + U [24576,256]  (~59 MB of fp32)
    float* ws = (float*)d_ws;
    const size_t GN = (size_t)NSMP * HDIM;
    float* g1 = ws;
    float* g2 = ws + 1 * GN;
    float* g3 = ws + 2 * GN;
    float* g4 = ws + 3 * GN;
    float* U  = ws + 4 * GN;   // 3*B x 256; forward h aliases its first third
    float* h  = U;

    const dim3 blk(256);
    const dim3 gF(NSMP / BM, 1);       // forward: 128 blocks, BN=256
    const dim3 gB(3 * NSMP / BM, 1);   // backward: 384 blocks

    // forward: pre_i = h @ W_i^T + b_i ; fused ELU / ELU' (in-place h is safe:
    // each block reads exactly the rows it rewrites, full N per block)
    gemm_wmma<0, 256><<<gF, blk, 0, stream>>>(x, 64,  W1, 64,  b1, nullptr, g1, h, nullptr, NSMP, HDIM, 64);
    gemm_wmma<0, 256><<<gF, blk, 0, stream>>>(h, 256, W2, 256, b2, nullptr, g2, h, nullptr, NSMP, HDIM, HDIM);
    gemm_wmma<0, 256><<<gF, blk, 0, stream>>>(h, 256, W3, 256, b3, nullptr, g3, h, nullptr, NSMP, HDIM, HDIM);
    gemm_wmma<1, 256><<<gF, blk, 0, stream>>>(h, 256, W4, 256, b4, nullptr, g4, nullptr, nullptr, NSMP, HDIM, HDIM);

    // U4 = rowbcast(W5) * g4, rows stacked as m = r*B + b
    init_u4<<<3 * NSMP * HDIM / 256, blk, 0, stream>>>(W5, g4, U);

    // backward: U <- (U @ W_i) * g_{i-1}   (in-place, full-N blocks)
    gemm_wmma<2, 256><<<gB, blk, 0, stream>>>(U, 256, W4, 256, nullptr, g3, nullptr, nullptr, U, 3 * NSMP, HDIM, HDIM);
    gemm_wmma<2, 256><<<gB, blk, 0, stream>>>(U, 256, W3, 256, nullptr, g2, nullptr, nullptr, U, 3 * NSMP, HDIM, HDIM);
    gemm_wmma<2, 256><<<gB, blk, 0, stream>>>(U, 256, W2, 256, nullptr, g1, nullptr, nullptr, U, 3 * NSMP, HDIM, HDIM);

    // final: out[b,r,:] = U1[r*B+b,:] @ W1   (N = 64)
    gemm_wmma<3, 64><<<gB, blk, 0, stream>>>(U, 256, W1, 64, nullptr, nullptr, nullptr, nullptr, out, 3 * NSMP, 64, HDIM);
}